// SingleStageDetector_82343112999573
// MI455X (gfx1250) — compile-verified
//
#include <hip/hip_runtime.h>

// ---------------------------------------------------------------------------
// SingleStageDetector IoU: decode anchors -> pairwise IoU vs gt boxes.
// B=128, A=9, H=W=7 (P=441 proposals), N=128 gt, out = (B, 441, 128) f32.
// Memory-bound: ~30 MB traffic => ~1.3 us floor @ 23.3 TB/s.
// CDNA5 path: async global->LDS staging of gt boxes (ASYNCcnt), NT stores.
// ---------------------------------------------------------------------------

#define GRID_SCALE 7.0f

constexpr int kA = 9, kH = 7, kW = 7, kN = 128;
constexpr int kP = kA * kH * kW;          // 441 proposals per batch
constexpr int kTileP = 112;               // proposals per block
constexpr int kTiles = (kP + kTileP - 1) / kTileP;  // 4
constexpr int kThreads = 256;             // 8 waves (wave32)

#if defined(__gfx1250__) && __has_builtin(__builtin_amdgcn_global_load_async_to_lds_b128)
#define USE_ASYNC_LDS 1
#else
#define USE_ASYNC_LDS 0
#endif

#if USE_ASYNC_LDS
typedef int v4i_t __attribute__((ext_vector_type(4)));
typedef __attribute__((address_space(1))) v4i_t as1_v4i;   // global
typedef __attribute__((address_space(3))) v4i_t as3_v4i;   // LDS
#endif

__global__ __launch_bounds__(kThreads)
void iou_kernel(const float* __restrict__ offsets,   // (B, A, H, W, 4)
                const float* __restrict__ bboxes,    // (B, N, 5)
                float* __restrict__ out)             // (B, P, N)
{
    __shared__ float sm_g[kN * 5];      // raw gt rows: x0,y0,x1,y1,cls
    __shared__ float sm_p[kTileP * 5];  // decoded proposals: x0,y0,x1,y1,area

    const int tid  = threadIdx.x;
    const int tile = blockIdx.x;
    const int b    = blockIdx.y;
    const int p0   = tile * kTileP;

    // ---- stage this batch's gt boxes (128*5 f32 = 2560 B = 160 x b128) ----
    const float* gsrc = bboxes + (size_t)b * (kN * 5);
#if USE_ASYNC_LDS
    if (tid < (kN * 5) / 4) {  // 160 lanes x 16 B
        __builtin_amdgcn_global_load_async_to_lds_b128(
            (as1_v4i*)(gsrc + tid * 4),
            (as3_v4i*)(sm_g + tid * 4),
            /*offset=*/0, /*cpol=*/0);
    }
#else
    if (tid < (kN * 5) / 4) {
        reinterpret_cast<float4*>(sm_g)[tid] =
            reinterpret_cast<const float4*>(gsrc)[tid];
    }
#endif

    // ---- decode this tile's proposals into LDS (contiguous float4 loads) ----
    if (tid < kTileP) {
        const int p = p0 + tid;
        if (p < kP) {
            float4 t = reinterpret_cast<const float4*>(offsets)[(size_t)b * kP + p];
            const int hw = p % (kH * kW);
            const float w = (float)(hw % kW);
            const float h = (float)(hw / kW);
            const float cx = (w + 0.5f) + t.x;
            const float cy = (h + 0.5f) + t.y;
            const float pw = t.z * GRID_SCALE;
            const float ph = t.w * GRID_SCALE;
            const float x0 = cx - 0.5f * pw, y0 = cy - 0.5f * ph;
            const float x1 = cx + 0.5f * pw, y1 = cy + 0.5f * ph;
            sm_p[tid * 5 + 0] = x0;
            sm_p[tid * 5 + 1] = y0;
            sm_p[tid * 5 + 2] = x1;
            sm_p[tid * 5 + 3] = y1;
            sm_p[tid * 5 + 4] = (x1 - x0) * (y1 - y0);  // match reference p_area
        }
    }

#if USE_ASYNC_LDS
#if __has_builtin(__builtin_amdgcn_s_wait_asynccnt)
    __builtin_amdgcn_s_wait_asynccnt(0);
#else
    asm volatile("s_wait_asynccnt 0" ::: "memory");
#endif
#endif
    __syncthreads();

    // ---- per-thread gt box lives in registers for the whole loop ----
    const int n = tid & (kN - 1);                 // lane-consecutive -> coalesced
    const float gx0 = sm_g[n * 5 + 0];            // stride-5 dwords: bank-conflict-free
    const float gy0 = sm_g[n * 5 + 1];
    const float gx1 = sm_g[n * 5 + 2];
    const float gy1 = sm_g[n * 5 + 3];
    const float garea = (gx1 - gx0) * (gy1 - gy0);

    float* outb = out + (size_t)b * kP * kN;

    // p_local uniform across each wave -> proposal LDS reads are broadcasts
    for (int pl = (tid >> 7); pl < kTileP; pl += kThreads / kN) {
        const int p = p0 + pl;
        if (p >= kP) break;
        const float px0 = sm_p[pl * 5 + 0];
        const float py0 = sm_p[pl * 5 + 1];
        const float px1 = sm_p[pl * 5 + 2];
        const float py1 = sm_p[pl * 5 + 3];
        const float parea = sm_p[pl * 5 + 4];

        const float xtl = fmaxf(px0, gx0);
        const float ytl = fmaxf(py0, gy0);
        const float xbr = fminf(px1, gx1);
        const float ybr = fminf(py1, gy1);
        const float iw = fmaxf(xbr - xtl, 0.0f);
        const float ih = fmaxf(ybr - ytl, 0.0f);
        const float inter = iw * ih;
        const float iou = inter / (parea + garea - inter);

        // write-once 29 MB output: stream past the caches
        __builtin_nontemporal_store(iou, outb + (size_t)p * kN + n);
    }
}

extern "C" void kernel_launch(void* const* d_in, const int* in_sizes, int n_in,
                              void* d_out, int out_size, void* d_ws, size_t ws_size,
                              hipStream_t stream) {
    const float* offsets = (const float*)d_in[0];  // (B, A, H, W, 4) f32
    const float* bboxes  = (const float*)d_in[1];  // (B, N, 5) f32
    float* out = (float*)d_out;                    // (B, P, N) f32

    const int B = in_sizes[0] / (kP * 4);          // robust to B changes
    dim3 grid(kTiles, B);
    iou_kernel<<<grid, dim3(kThreads), 0, stream>>>(offsets, bboxes, out);
}